// Mamba_1812476199131
// MI455X (gfx1250) — compile-verified
//
#include <hip/hip_runtime.h>
#include <hip/hip_bf16.h>
#include <math.h>

// ---------------------------------------------------------------------------
// Mamba block for MI455X (gfx1250, wave32, WMMA + TDM).
//   in_proj / out_proj : WMMA f32 GEMM, W panel staged into LDS by the Tensor
//                        Data Mover (tensor_load_to_lds + s_wait_tensorcnt),
//                        double-buffered; A panel register-double-buffered.
//   x_proj / dt_proj   : direct-global WMMA f32 GEMM (small).
//   conv+SiLU, scan    : elementwise / wave-parallel recurrence.
// ---------------------------------------------------------------------------

#define MB 2
#define ML 2048
#define MDM 768
#define MDI 1536           // 2*DM
#define MN 16              // d_state
#define MR 48              // dt_rank
#define MK 4               // d_conv
#define MROWS (MB * ML)    // 4096 flattened (b, t) rows

typedef __attribute__((ext_vector_type(2))) float v2f;
typedef __attribute__((ext_vector_type(8))) float v8f;
typedef __attribute__((ext_vector_type(4))) unsigned int v4u;
typedef __attribute__((ext_vector_type(4))) int v4i;
typedef __attribute__((ext_vector_type(8))) int v8i;

#define KC 64                       // k-chunk per TDM stage (16 k-steps)
#define WROW 66                     // LDS row stride in floats (64 data + 2 pad)
#define WBUF_BYTES (64 * WROW * 4)  // one staged W panel: 16896 B

// ---------------------------------------------------------------------------
// TDM: DMA a 64-row x 64-element f32 tile of W (row-major, row stride ldb)
// into LDS at byte offset lds_off.  Padding: +2 DWORDs after every 64 DWORDs
// -> LDS row stride 66 floats (bank-conflict-free fragment reads).
// ---------------------------------------------------------------------------
__device__ __forceinline__ void tdm_load_w_tile(const float* gsrc,
                                                unsigned lds_off,
                                                int Kdim, int Ndim, int ldb) {
  unsigned long long ga = (unsigned long long)(uintptr_t)gsrc;
  v4u g0;
  g0.x = 1u;                                          // count=1, user mode
  g0.y = lds_off;                                     // lds_addr (bytes)
  g0.z = (unsigned)(ga & 0xffffffffu);                // global_addr[31:0]
  g0.w = (unsigned)((ga >> 32) & 0x01ffffffu)         // global_addr[56:32]
         | (2u << 30);                                // type = 2 ("image")

  v8i g1;
  g1[0] = (int)((2u << 16)      // data_size = 4B
                | (1u << 20)    // pad_enable
                | (5u << 22)    // pad_interval: 64 DWORDs
                | (1u << 25));  // pad_amount: 2 DWORDs
  g1[1] = (int)(((unsigned)Kdim & 0xffffu) << 16);                    // tensor_dim0 lo
  g1[2] = (int)(((unsigned)Kdim >> 16) | (((unsigned)Ndim & 0xffffu) << 16));
  g1[3] = (int)(((unsigned)Ndim >> 16) | ((unsigned)KC << 16));       // tile_dim0 = 64
  g1[4] = (int)64u;                                                   // tile_dim1 = 64 rows
  g1[5] = (int)(unsigned)ldb;                                         // tensor_dim0_stride lo
  g1[6] = 0;
  g1[7] = 0;

  v4i g2 = {0, 0, 0, 0};
  v4i g3 = {0, 0, 0, 0};
#if __clang_major__ >= 23
  v8i g4 = {0, 0, 0, 0, 0, 0, 0, 0};
  __builtin_amdgcn_tensor_load_to_lds(g0, g1, g2, g3, g4, 0);
#else
  __builtin_amdgcn_tensor_load_to_lds(g0, g1, g2, g3, 0);
#endif
}

// ---------------------------------------------------------------------------
// TDM-staged WMMA GEMM:  C[M,N] = A[M,K] * W[N,K]^T
// Block = 4 waves -> 64x64 tile.  W panel double-buffered in LDS via TDM;
// A panel double-buffered in registers (16 x b64 clause per chunk).
// ---------------------------------------------------------------------------
__global__ __launch_bounds__(128) void gemm_wmma_tdm(
    const float* __restrict__ A, const float* __restrict__ W,
    float* __restrict__ C, int Ndim, int Kdim, int lda, int ldb, int ldc) {
  __shared__ float smW[2][64 * WROW];

  // --- Escape smW's address + opaque store: the TDM builtin writes LDS via a
  // descriptor (no pointer), so without these, LLVM proves smW is never
  // stored / never escapes and folds the ds_loads to undef (seen in r2/r3).
  float* esc = &smW[0][0];
  int opaque0;
  asm volatile("s_mov_b32 %0, 0" : "=s"(opaque0));
  if (opaque0) smW[0][0] = 0.0f;               // never executes at runtime
  // Handoff fence: asm may read/write memory through the escaped pointer.
#define TDM_LDS_FENCE() asm volatile("" : "+v"(esc) :: "memory")

  const int lane = threadIdx.x;         // 0..31
  const int wave = threadIdx.y;         // 0..3
  const int half = lane >> 4;
  const int l16  = lane & 15;
  const int m0 = (blockIdx.y * 4 + wave) * 16;
  const int n0 = blockIdx.x * 64;

  const float* arow = A + (size_t)(m0 + l16) * lda + half * 2;
  const float* wpan = W + (size_t)n0 * ldb;

  v8f acc[4];
#pragma unroll
  for (int t = 0; t < 4; ++t) acc[t] = (v8f){0.f, 0.f, 0.f, 0.f, 0.f, 0.f, 0.f, 0.f};

  const int nch = Kdim / KC;

  // ---- prologue: stage W chunk 0 (TDM) and A chunk 0 (registers) ----
  v2f a_cur[16], a_nxt[16];
#pragma unroll
  for (int i = 0; i < 16; ++i) a_cur[i] = *(const v2f*)(arow + i * 4);

  if (wave == 0) {
    tdm_load_w_tile(wpan, 0u, Kdim, Ndim, ldb);
    __builtin_amdgcn_s_wait_tensorcnt(0);
  }
  __syncthreads();

  int cur = 0;
  for (int ch = 0; ch < nch; ++ch) {
    const bool more = (ch + 1 < nch);
    const int kb_next = (ch + 1) * KC;

    // Issue next chunk's DMA + A-loads before computing current chunk.
    if (wave == 0 && more)
      tdm_load_w_tile(wpan + kb_next, (unsigned)((cur ^ 1) * WBUF_BYTES),
                      Kdim, Ndim, ldb);
    if (more) {
      const float* an = arow + kb_next;
#pragma unroll
      for (int i = 0; i < 16; ++i) a_nxt[i] = *(const v2f*)(an + i * 4);
    }

    TDM_LDS_FENCE();
#pragma unroll
    for (int kk4 = 0; kk4 < 16; ++kk4) {
#pragma unroll
      for (int t = 0; t < 4; ++t) {
        v2f b = *(const v2f*)&smW[cur][(t * 16 + l16) * WROW + kk4 * 4 + half * 2];
        acc[t] = __builtin_amdgcn_wmma_f32_16x16x4_f32(
            false, a_cur[kk4], false, b, (short)0, acc[t], false, false);
      }
    }
    TDM_LDS_FENCE();

    if (wave == 0 && more) __builtin_amdgcn_s_wait_tensorcnt(0);
    __syncthreads();
    cur ^= 1;
#pragma unroll
    for (int i = 0; i < 16; ++i) a_cur[i] = a_nxt[i];
  }

#pragma unroll
  for (int t = 0; t < 4; ++t) {
    const int n = n0 + t * 16 + l16;
#pragma unroll
    for (int v = 0; v < 8; ++v) {
      const int m = m0 + half * 8 + v;
      C[(size_t)m * ldc + n] = acc[t][v];
    }
  }
#undef TDM_LDS_FENCE
}

// ---------------------------------------------------------------------------
// Direct-global WMMA f32 GEMM (small problems):
//   C[M,N] = act( A[M,K] * W[N,K]^T (+ bias) ),  ACT: 0 none, 1 bias+softplus
// ---------------------------------------------------------------------------
template <int NT, int ACT>
__global__ __launch_bounds__(128) void gemm_wmma_f32(
    const float* __restrict__ A, const float* __restrict__ W,
    const float* __restrict__ bias, float* __restrict__ C,
    int Kdim, int lda, int ldb, int ldc) {
  const int lane = threadIdx.x;
  const int wave = threadIdx.y;
  const int half = lane >> 4;
  const int l16  = lane & 15;

  const int m0 = (blockIdx.y * 4 + wave) * 16;
  const int n0 = blockIdx.x * (16 * NT);

  const float* arow = A + (size_t)(m0 + l16) * lda + half * 2;
  const float* wrow[NT];
#pragma unroll
  for (int t = 0; t < NT; ++t)
    wrow[t] = W + (size_t)(n0 + t * 16 + l16) * ldb + half * 2;

  v8f acc[NT];
#pragma unroll
  for (int t = 0; t < NT; ++t) acc[t] = (v8f){0.f, 0.f, 0.f, 0.f, 0.f, 0.f, 0.f, 0.f};

  for (int k = 0; k < Kdim; k += 4) {
    v2f a = *(const v2f*)(arow + k);
#pragma unroll
    for (int t = 0; t < NT; ++t) {
      v2f b = *(const v2f*)(wrow[t] + k);
      acc[t] = __builtin_amdgcn_wmma_f32_16x16x4_f32(
          false, a, false, b, (short)0, acc[t], false, false);
    }
  }

#pragma unroll
  for (int t = 0; t < NT; ++t) {
    const int n = n0 + t * 16 + l16;
#pragma unroll
    for (int v = 0; v < 8; ++v) {
      const int m = m0 + half * 8 + v;
      float val = acc[t][v];
      if constexpr (ACT == 1) {
        val += bias[n];
        val = (val > 20.0f) ? val : log1pf(__expf(val));   // softplus
      }
      C[(size_t)m * ldc + n] = val;
    }
  }
}

// ---------------------------------------------------------------------------
// Causal depthwise conv1d (K=4) + SiLU.  Reads x-half of xz, writes xconv.
// ---------------------------------------------------------------------------
__global__ __launch_bounds__(256) void conv_silu_kernel(
    const float* __restrict__ xz, const float* __restrict__ conv_w,
    const float* __restrict__ conv_b, float* __restrict__ xconv) {
  const int idx = blockIdx.x * 256 + threadIdx.x;
  const int total = MB * ML * MDI;
  if (idx >= total) return;
  const int d = idx % MDI;
  const int t = (idx / MDI) % ML;
  const int b = idx / (MDI * ML);

  float acc = conv_b[d];
  const float* wv = conv_w + d * MK;
  const size_t rowbase = (size_t)(b * ML) * (2 * MDI) + d;
#pragma unroll
  for (int j = 0; j < MK; ++j) {
    const int tt = t - (MK - 1) + j;
    if (tt >= 0) acc += wv[j] * xz[rowbase + (size_t)tt * (2 * MDI)];
  }
  acc = acc / (1.0f + __expf(-acc));   // SiLU
  xconv[idx] = acc;
}

// ---------------------------------------------------------------------------
// Selective scan.  One thread = one (b, d, n) state; 16 lanes per channel d,
// two channels per wave32.  h_t = exp(delta*A)*h + (delta*x)*B, y = sum_n h*C.
// shfl_xor masks 1/2/4/8 reduce within each 16-lane half.  Gate + D*x fused;
// result written in place over the xconv buffer (wave-lockstep safe).
// ---------------------------------------------------------------------------
__global__ __launch_bounds__(256) void scan_kernel(
    const float* __restrict__ delta, const float* __restrict__ xdbl,
    const float* __restrict__ A_log, const float* __restrict__ Dvec,
    const float* __restrict__ xz, float* __restrict__ xy) {
  const int idx = blockIdx.x * 256 + threadIdx.x;   // exact: 192*256 = B*DI*16
  const int n = idx & 15;
  const int d = (idx >> 4) % MDI;
  const int b = idx / (16 * MDI);

  const float Adn = -__expf(A_log[d * MN + n]);
  const float Dd  = Dvec[d];
  float h = 0.0f;
  const size_t baseBL = (size_t)b * ML;

  for (int t = 0; t < ML; ++t) {
    const size_t row = baseBL + t;
    const float dl = delta[row * MDI + d];
    const float xv = xy[row * MDI + d];
    const float Bv = xdbl[row * 80 + MR + n];
    const float Cv = xdbl[row * 80 + MR + MN + n];

    h = __expf(dl * Adn) * h + (dl * xv) * Bv;

    float p = h * Cv;
    p += __shfl_xor(p, 1);
    p += __shfl_xor(p, 2);
    p += __shfl_xor(p, 4);
    p += __shfl_xor(p, 8);

    if (n == 0) {
      float y = p + Dd * xv;
      const float zv = xz[row * (2 * MDI) + MDI + d];
      y *= zv / (1.0f + __expf(-zv));   // * silu(z)
      xy[row * MDI + d] = y;
    }
  }
}

// ---------------------------------------------------------------------------
extern "C" void kernel_launch(void* const* d_in, const int* in_sizes, int n_in,
                              void* d_out, int out_size, void* d_ws, size_t ws_size,
                              hipStream_t stream) {
  const float* hs        = (const float*)d_in[0];
  const float* in_proj_w = (const float*)d_in[1];
  const float* conv_w    = (const float*)d_in[2];
  const float* conv_b    = (const float*)d_in[3];
  const float* x_proj_w  = (const float*)d_in[4];
  const float* dt_proj_w = (const float*)d_in[5];
  const float* dt_proj_b = (const float*)d_in[6];
  const float* A_log     = (const float*)d_in[7];
  const float* Dvec      = (const float*)d_in[8];
  const float* out_proj_w= (const float*)d_in[9];
  float* out = (float*)d_out;

  // Workspace layout (floats), ~102 MB total
  float* ws    = (float*)d_ws;
  float* xz    = ws;                                   // 4096*3072
  float* xdbl  = xz   + (size_t)MROWS * 2 * MDI;       // 4096*80
  float* delta = xdbl + (size_t)MROWS * 80;            // 4096*1536
  float* xy    = delta+ (size_t)MROWS * MDI;           // 4096*1536

  const dim3 blk(32, 4);

  // 1) xz = hs @ in_proj_w^T    (M=4096, N=3072, K=768)  [TDM + WMMA]
  gemm_wmma_tdm<<<dim3(2 * MDI / 64, MROWS / 64), blk, 0, stream>>>(
      hs, in_proj_w, xz, 2 * MDI, MDM, MDM, MDM, 2 * MDI);

  // 2) depthwise causal conv + SiLU -> xy (holds x for now)
  conv_silu_kernel<<<(MB * ML * MDI) / 256, 256, 0, stream>>>(xz, conv_w, conv_b, xy);

  // 3) x_dbl = x @ x_proj_w^T   (M=4096, N=80, K=1536)
  gemm_wmma_f32<1, 0><<<dim3(80 / 16, MROWS / 64), blk, 0, stream>>>(
      xy, x_proj_w, nullptr, xdbl, MDI, MDI, MDI, 80);

  // 4) delta = softplus(dt @ dt_proj_w^T + b)   (M=4096, N=1536, K=48, lda=80)
  gemm_wmma_f32<4, 1><<<dim3(MDI / 64, MROWS / 64), blk, 0, stream>>>(
      xdbl, dt_proj_w, dt_proj_b, delta, MR, 80, MR, MDI);

  // 5) selective scan + gate, in place into xy
  scan_kernel<<<(MB * MDI * MN) / 256, 256, 0, stream>>>(delta, xdbl, A_log, Dvec, xz, xy);

  // 6) out = y @ out_proj_w^T   (M=4096, N=768, K=1536)  [TDM + WMMA]
  gemm_wmma_tdm<<<dim3(MDM / 64, MROWS / 64), blk, 0, stream>>>(
      xy, out_proj_w, out, MDM, MDI, MDI, MDI, MDM);
}